// DiagonalUpsample_81862076662185
// MI455X (gfx1250) — compile-verified
//
#include <hip/hip_runtime.h>

// DiagonalUpsample: out[b,c,2h,2w+?] checkerboard-interleave of up/down.
// Pure streaming kernel: 256 MiB in, 512 MiB out, 0 FLOPs -> HBM-bound,
// floor ~35 us at 23.3 TB/s. Optimized for b128 NT vmem traffic, wave32.

typedef __attribute__((ext_vector_type(4))) float v4f;

// Fixed problem geometry from the reference (powers of two -> shift/mask math).
#define DU_W 256
#define DU_H 256
#define DU_WQ (DU_W / 4)   // quads per input row = 64

__global__ __launch_bounds__(256) void DiagonalUpsample_kernel(
    const float* __restrict__ up,
    const float* __restrict__ down,
    float* __restrict__ out)
{
    // One thread per 4 consecutive input floats along W.
    const unsigned tid = blockIdx.x * 256u + threadIdx.x;

    const unsigned wq  = tid & (DU_WQ - 1);   // quad index along W   [0,64)
    const unsigned row = tid >> 6;            // flattened (b,c,h) row
    const unsigned h   = row & (DU_H - 1);    // h within image       [0,256)
    const unsigned bc  = row >> 8;            // flattened (b,c)

    // ---- load 4 floats from each source (b128, non-temporal: read-once) ----
    const size_t in_off = (size_t)row * DU_W + (size_t)wq * 4;
    const v4f u = __builtin_nontemporal_load((const v4f*)(up   + in_off));
    const v4f d = __builtin_nontemporal_load((const v4f*)(down + in_off));

    // ---- interleave into two output rows of 8 floats each ----
    // top row (2h):    d0 u0 d1 u1 | d2 u2 d3 u3
    // bot row (2h+1):  u0 d0 u1 d1 | u2 d2 u3 d3
    v4f t0 = (v4f){ d.x, u.x, d.y, u.y };
    v4f t1 = (v4f){ d.z, u.z, d.w, u.w };
    v4f b0 = (v4f){ u.x, d.x, u.y, d.y };
    v4f b1 = (v4f){ u.z, d.z, u.w, d.w };

    // ---- store (4x b128, non-temporal: 512 MiB stream >> 192 MB L2) ----
    const size_t orow0 = ((size_t)bc * (2 * DU_H) + 2 * (size_t)h) * (2 * DU_W)
                       + (size_t)wq * 8;
    float* o0 = out + orow0;            // row 2h
    float* o1 = o0 + 2 * DU_W;          // row 2h+1

    __builtin_nontemporal_store(t0, (v4f*)(o0));
    __builtin_nontemporal_store(t1, (v4f*)(o0 + 4));
    __builtin_nontemporal_store(b0, (v4f*)(o1));
    __builtin_nontemporal_store(b1, (v4f*)(o1 + 4));
}

extern "C" void kernel_launch(void* const* d_in, const int* in_sizes, int n_in,
                              void* d_out, int out_size, void* d_ws, size_t ws_size,
                              hipStream_t stream) {
    const float* up   = (const float*)d_in[0];   // up_diagonal
    const float* down = (const float*)d_in[1];   // down_diagonal
    float* out = (float*)d_out;

    // in_sizes[0] = B*C*H*W = 33,554,432; one thread per 4 elements.
    const unsigned total_threads = (unsigned)(in_sizes[0] / 4);   // 8,388,608
    const unsigned block = 256;                                   // 8 wave32s
    const unsigned grid  = total_threads / block;                 // 32,768

    DiagonalUpsample_kernel<<<grid, block, 0, stream>>>(up, down, out);
}